// GAT_45260365365956
// MI455X (gfx1250) — compile-verified
//
#include <hip/hip_runtime.h>
#include <hip/hip_bf16.h>

// GATv2 3-layer GNN inference for MI455X (gfx1250).
//  - Dense node transforms: v_wmma_f32_16x16x4_f32, weights staged in LDS by
//    the Tensor Data Mover (tensor_load_to_lds + s_wait_tensorcnt).
//  - Edge softmax/aggregation: wave32-per-edge streaming passes; gathers are
//    L2-resident (xl/xr = 25.6 MB each << 192 MB L2).

typedef float v2f  __attribute__((ext_vector_type(2)));
typedef float v8f  __attribute__((ext_vector_type(8)));
typedef unsigned int u32x4 __attribute__((ext_vector_type(4)));
typedef int   i32x4 __attribute__((ext_vector_type(4)));
typedef int   i32x8 __attribute__((ext_vector_type(8)));

#define NEG_SLOPE 0.2f
#define BN_EPS 1e-5f

__device__ __forceinline__ float lrelu(float x) {
    return x > 0.f ? x : NEG_SLOPE * x;
}

// Float atomic max via int punning: non-negative -> signed int max,
// negative -> unsigned int min. Slots must be initialized to -inf bits.
__device__ __forceinline__ void atomicMaxF(float* addr, float val) {
    int iv = __float_as_int(val);
    if (iv >= 0) atomicMax((int*)addr, iv);
    else         atomicMin((unsigned int*)addr, (unsigned int)iv);
}

// ---------------------------------------------------------------------------
// C[M x NC] = A[M x K] @ B[K x NC] (+ bias). One wave per 16x16 C tile,
// persistent blocks loop over tiles. B (<=64KB) staged in LDS via TDM.
// Requires: M % 16 == 0, K % 4 == 0 (true here: M=50000, K in {100,128}).
// WMMA f32 16x16x4 operand layout (ISA 7.12.2):
//   A (16x4): lanes 0-15 rows M=0..15; VGPR0/1 = K(+0)/K(+1) low half-wave,
//             K(+2)/K(+3) high half-wave.
//   B (4x16): rows striped across lanes; low half K(+0)/K(+1), high half
//             K(+2)/K(+3).
//   C/D: VGPR j -> M=j (lanes 0-15, N=lane) / M=j+8 (lanes 16-31, N=lane-16).
// ---------------------------------------------------------------------------
template <int K, int NC>
__global__ void wmma_gemm_tdm(const float* __restrict__ A,
                              const float* __restrict__ B,
                              const float* __restrict__ bias,
                              float* __restrict__ C,
                              int M)
{
    __shared__ float ldsB[K * NC];   // only LDS object -> LDS offset 0

#if __has_builtin(__builtin_amdgcn_tensor_load_to_lds)
    if (threadIdx.x < 32) {
        // Tensor DMA descriptor: 2D tile K x NC of 4-byte elements,
        // tile == tensor, row stride NC, destination LDS offset 0.
        const unsigned long long ga = (unsigned long long)(size_t)B;
        u32x4 g0;
        g0[0] = 1u;                                    // count=1 (valid user D#)
        g0[1] = 0u;                                    // lds_addr = 0
        g0[2] = (unsigned)(ga & 0xFFFFFFFFu);          // global_addr[31:0]
        g0[3] = (unsigned)((ga >> 32) & 0x01FFFFFFu)   // global_addr[56:32]
              | (2u << 30);                            // type = 2 ("image")
        i32x8 g1;
        g1[0] = (2 << 16);                             // data_size = 4B
        g1[1] = (NC & 0xFFFF) << 16;                   // tensor_dim0[15:0]
        g1[2] = ((NC >> 16) & 0xFFFF)                  // tensor_dim0[31:16]
              | ((K & 0xFFFF) << 16);                  // tensor_dim1[15:0]
        g1[3] = ((K >> 16) & 0xFFFF)                   // tensor_dim1[31:16]
              | ((NC & 0xFFFF) << 16);                 // tile_dim0 = NC
        g1[4] = (K & 0xFFFF);                          // tile_dim1 = K (dim2=0)
        g1[5] = NC;                                    // tensor_dim0_stride lo
        g1[6] = 0;
        g1[7] = 0;
        i32x4 z4 = {0, 0, 0, 0};
#if __clang_major__ >= 23
        i32x8 z8 = {0, 0, 0, 0, 0, 0, 0, 0};
        __builtin_amdgcn_tensor_load_to_lds(g0, g1, z4, z4, z8, 0);
#else
        __builtin_amdgcn_tensor_load_to_lds(g0, g1, z4, z4, 0);
#endif
        __builtin_amdgcn_s_wait_tensorcnt(0);
    }
#else
    for (int i = threadIdx.x; i < K * NC; i += blockDim.x) ldsB[i] = B[i];
#endif
    __syncthreads();

    const int wpb    = blockDim.x >> 5;
    const int wid    = threadIdx.x >> 5;
    const int lane   = threadIdx.x & 31;
    const int half   = lane >> 4;          // 0: lanes 0-15, 1: lanes 16-31
    const int l      = lane & 15;
    const int tilesN = (NC + 15) >> 4;
    const int tilesM = M >> 4;
    const int total  = tilesM * tilesN;

    for (int tile = blockIdx.x * wpb + wid; tile < total;
         tile += gridDim.x * wpb) {
        const int tm = tile / tilesN;
        const int tn = tile % tilesN;
        const int rowA = tm * 16 + l;                  // always < M
        const int colB = tn * 16 + l;
        const int colC = (colB < NC) ? colB : (NC - 1);   // clamp (branch-free)

        const float* __restrict__ ap = A + (size_t)rowA * K;
        v8f acc = {0.f, 0.f, 0.f, 0.f, 0.f, 0.f, 0.f, 0.f};
#pragma unroll
        for (int k0 = 0; k0 < K; k0 += 4) {
            const int ka = k0 + 2 * half;              // even -> 8B aligned
            const v2f a = *(const v2f*)(ap + ka);
            v2f b;
            b.x = ldsB[ka * NC + colC];
            b.y = ldsB[(ka + 1) * NC + colC];
            acc = __builtin_amdgcn_wmma_f32_16x16x4_f32(
                false, a, false, b, (short)0, acc, false, false);
        }
        if (colB < NC) {
            const float bv = bias ? bias[colB] : 0.f;
#pragma unroll
            for (int j = 0; j < 8; ++j) {
                const int m = tm * 16 + j + 8 * half;  // always < M
                C[(size_t)m * NC + colB] = acc[j] + bv;
            }
        }
    }
}

// ---------------------------------------------------------------------------
// Edge pass 1 (heads=4, ch=32, F=128): per-edge attention logits + seg-max.
// One wave32 per edge; lane covers channel `lane` of each of the 4 heads.
// ---------------------------------------------------------------------------
__global__ void edge_logits4(const float* __restrict__ xl,
                             const float* __restrict__ xr,
                             const int* __restrict__ srcI,
                             const int* __restrict__ dstI,
                             const float* __restrict__ att,
                             float* __restrict__ etmp,
                             float* __restrict__ nmax,
                             int Ereal, int Etot)
{
    const int wave = blockIdx.x * (blockDim.x >> 5) + (threadIdx.x >> 5);
    if (wave >= Etot) return;
    const int lane = threadIdx.x & 31;
    int s, d;
    if (wave < Ereal) { s = srcI[wave]; d = dstI[wave]; }
    else              { s = d = wave - Ereal; }  // appended self-loop

    const float* pl = xl + (size_t)s * 128;
    const float* pr = xr + (size_t)d * 128;
    float h0 = lrelu(pl[lane     ] + pr[lane     ]) * att[lane     ];
    float h1 = lrelu(pl[lane + 32] + pr[lane + 32]) * att[lane + 32];
    float h2 = lrelu(pl[lane + 64] + pr[lane + 64]) * att[lane + 64];
    float h3 = lrelu(pl[lane + 96] + pr[lane + 96]) * att[lane + 96];
#pragma unroll
    for (int off = 16; off > 0; off >>= 1) {
        h0 += __shfl_xor(h0, off, 32);
        h1 += __shfl_xor(h1, off, 32);
        h2 += __shfl_xor(h2, off, 32);
        h3 += __shfl_xor(h3, off, 32);
    }
    if (lane == 0) {
        float* et = etmp + (size_t)wave * 4;
        et[0] = h0; et[1] = h1; et[2] = h2; et[3] = h3;
        float* nm = nmax + (size_t)d * 4;
        atomicMaxF(&nm[0], h0);
        atomicMaxF(&nm[1], h1);
        atomicMaxF(&nm[2], h2);
        atomicMaxF(&nm[3], h3);
    }
}

// Edge pass 1 for heads=1 (layer 3, F=47).
__global__ void edge_logits1(const float* __restrict__ xl,
                             const float* __restrict__ xr,
                             const int* __restrict__ srcI,
                             const int* __restrict__ dstI,
                             const float* __restrict__ att,
                             float* __restrict__ etmp,
                             float* __restrict__ nmax,
                             int Ereal, int Etot, int F)
{
    const int wave = blockIdx.x * (blockDim.x >> 5) + (threadIdx.x >> 5);
    if (wave >= Etot) return;
    const int lane = threadIdx.x & 31;
    int s, d;
    if (wave < Ereal) { s = srcI[wave]; d = dstI[wave]; }
    else              { s = d = wave - Ereal; }

    const float* pl = xl + (size_t)s * F;
    const float* pr = xr + (size_t)d * F;
    float a = 0.f;
    if (lane < F)      a  = lrelu(pl[lane]      + pr[lane]     ) * att[lane];
    if (lane + 32 < F) a += lrelu(pl[lane + 32] + pr[lane + 32]) * att[lane + 32];
#pragma unroll
    for (int off = 16; off > 0; off >>= 1) a += __shfl_xor(a, off, 32);
    if (lane == 0) {
        etmp[wave] = a;
        atomicMaxF(&nmax[d], a);
    }
}

// Edge pass 2: ex = exp(e - max[dst]); seg-sum into denom. Thread per (edge,head).
__global__ void edge_exp(const int* __restrict__ dstI,
                         float* __restrict__ etmp,
                         const float* __restrict__ nmax,
                         float* __restrict__ nden,
                         int Ereal, int Etot, int heads)
{
    const int idx = blockIdx.x * blockDim.x + threadIdx.x;
    if (idx >= Etot * heads) return;
    const int e = idx / heads;
    const int h = idx - e * heads;
    const int d = (e < Ereal) ? dstI[e] : (e - Ereal);
    const float ex = __expf(etmp[idx] - nmax[(size_t)d * heads + h]);
    etmp[idx] = ex;
    atomicAdd(&nden[(size_t)d * heads + h], ex);
}

// Edge pass 3: agg[dst] += alpha * xl[src]. One wave32 per edge.
__global__ void edge_agg(const float* __restrict__ xl,
                         const int* __restrict__ srcI,
                         const int* __restrict__ dstI,
                         const float* __restrict__ etmp,
                         const float* __restrict__ nden,
                         float* __restrict__ agg,
                         int Ereal, int Etot, int F, int heads)
{
    const int wave = blockIdx.x * (blockDim.x >> 5) + (threadIdx.x >> 5);
    if (wave >= Etot) return;
    const int lane = threadIdx.x & 31;
    int s, d;
    if (wave < Ereal) { s = srcI[wave]; d = dstI[wave]; }
    else              { s = d = wave - Ereal; }

    float al[4];
#pragma unroll
    for (int h = 0; h < 4; ++h)
        al[h] = (h < heads)
              ? etmp[(size_t)wave * heads + h] / nden[(size_t)d * heads + h]
              : 0.f;

    const float* pl = xl + (size_t)s * F;
    float* po = agg + (size_t)d * F;
#pragma unroll
    for (int j = 0; j < 4; ++j) {
        const int f = lane + 32 * j;
        if (f < F) {
            const float alpha = (heads == 4) ? al[j] : al[0];
            atomicAdd(&po[f], alpha * pl[f]);
        }
    }
}

__global__ void init_softmax(float* __restrict__ nmax, float* __restrict__ nden, int n)
{
    const int i = blockIdx.x * blockDim.x + threadIdx.x;
    if (i < n) { nmax[i] = __int_as_float((int)0xFF800000); nden[i] = 0.f; }
}

__global__ void copy_f32(float* __restrict__ dst, const float* __restrict__ src, int n)
{
    const int i = blockIdx.x * blockDim.x + threadIdx.x;
    if (i < n) dst[i] = src[i];
}

// out = [relu]( BN( agg + gat_bias ) ); BN in eval mode.
__global__ void combine_bn(const float* __restrict__ agg,
                           const float* __restrict__ gbias,
                           const float* __restrict__ g,
                           const float* __restrict__ b,
                           const float* __restrict__ m,
                           const float* __restrict__ v,
                           float* __restrict__ out,
                           int total, int F, int doRelu)
{
    const int idx = blockIdx.x * blockDim.x + threadIdx.x;
    if (idx >= total) return;
    const int f = idx % F;
    float val = agg[idx] + gbias[f];
    val = (val - m[f]) * rsqrtf(v[f] + BN_EPS) * g[f] + b[f];
    if (doRelu) val = fmaxf(val, 0.f);
    out[idx] = val;
}

// ---------------------------------------------------------------------------
extern "C" void kernel_launch(void* const* d_in, const int* in_sizes, int n_in,
                              void* d_out, int out_size, void* d_ws, size_t ws_size,
                              hipStream_t stream)
{
    const int N = 50000, E = 800000, Etot = E + N;
    const int W1F = 128, OUTF = 47;

    const float* x     = (const float*)d_in[0];
    const int*   eidx  = (const int*)d_in[1];
    const int*   srcI  = eidx;
    const int*   dstI  = eidx + E;
    const float* w1s   = (const float*)d_in[2];
    const float* w1d   = (const float*)d_in[3];
    const float* att1  = (const float*)d_in[4];
    const float* b1    = (const float*)d_in[5];
    const float* bn1g  = (const float*)d_in[6];
    const float* bn1b  = (const float*)d_in[7];
    const float* bn1m  = (const float*)d_in[8];
    const float* bn1v  = (const float*)d_in[9];
    const float* res0w = (const float*)d_in[10];
    const float* res0b = (const float*)d_in[11];
    const float* w2s   = (const float*)d_in[12];
    const float* w2d   = (const float*)d_in[13];
    const float* att2  = (const float*)d_in[14];
    const float* b2    = (const float*)d_in[15];
    const float* bn2g  = (const float*)d_in[16];
    const float* bn2b  = (const float*)d_in[17];
    const float* bn2m  = (const float*)d_in[18];
    const float* bn2v  = (const float*)d_in[19];
    const float* w3s   = (const float*)d_in[20];
    const float* w3d   = (const float*)d_in[21];
    const float* att3  = (const float*)d_in[22];
    const float* b3    = (const float*)d_in[23];
    const float* bn3g  = (const float*)d_in[24];
    const float* bn3b  = (const float*)d_in[25];
    const float* bn3m  = (const float*)d_in[26];
    const float* bn3v  = (const float*)d_in[27];
    const float* res2w = (const float*)d_in[28];
    const float* res2b = (const float*)d_in[29];

    // Workspace layout (floats): ~118 MB total.
    float* XL  = (float*)d_ws;             // node src-transform  [N x 128]
    float* XR  = XL  + (size_t)N * 128;    // node dst-transform  [N x 128]
    float* AGG = XR  + (size_t)N * 128;    // aggregation acc     [N x 128]
    float* H   = AGG + (size_t)N * 128;    // layer activations   [N x 128]
    float* ET  = H   + (size_t)N * 128;    // per-edge logits/ex  [Etot x 4]
    float* NMX = ET  + (size_t)Etot * 4;   // seg-max             [N x 4]
    float* NDN = NMX + (size_t)N * 4;      // seg-sum             [N x 4]

    const int edgeBlocks = (Etot + 7) / 8;            // 8 waves / block
    const dim3 B256(256);
    const dim3 B128(128);                             // 4 waves / block
    const dim3 GEMM_GRID(1024);                       // persistent blocks

    // ---------------- Layer 1: 100 -> 4x32, linear residual ----------------
    hipLaunchKernelGGL((wmma_gemm_tdm<100, 128>), GEMM_GRID, B128, 0, stream,
                       x, w1s, (const float*)nullptr, XL, N);
    hipLaunchKernelGGL((wmma_gemm_tdm<100, 128>), GEMM_GRID, B128, 0, stream,
                       x, w1d, (const float*)nullptr, XR, N);
    hipLaunchKernelGGL((wmma_gemm_tdm<100, 128>), GEMM_GRID, B128, 0, stream,
                       x, res0w, res0b, AGG, N);      // AGG = residual
    hipLaunchKernelGGL(init_softmax, dim3((N * 4 + 255) / 256), B256, 0, stream,
                       NMX, NDN, N * 4);
    hipLaunchKernelGGL(edge_logits4, dim3(edgeBlocks), B256, 0, stream,
                       XL, XR, srcI, dstI, att1, ET, NMX, E, Etot);
    hipLaunchKernelGGL(edge_exp, dim3((Etot * 4 + 255) / 256), B256, 0, stream,
                       dstI, ET, NMX, NDN, E, Etot, 4);
    hipLaunchKernelGGL(edge_agg, dim3(edgeBlocks), B256, 0, stream,
                       XL, srcI, dstI, ET, NDN, AGG, E, Etot, W1F, 4);
    hipLaunchKernelGGL(combine_bn, dim3((N * W1F + 255) / 256), B256, 0, stream,
                       AGG, b1, bn1g, bn1b, bn1m, bn1v, H, N * W1F, W1F, 1);

    // ---------------- Layer 2: 128 -> 4x32, identity residual --------------
    hipLaunchKernelGGL((wmma_gemm_tdm<128, 128>), GEMM_GRID, B128, 0, stream,
                       H, w2s, (const float*)nullptr, XL, N);
    hipLaunchKernelGGL((wmma_gemm_tdm<128, 128>), GEMM_GRID, B128, 0, stream,
                       H, w2d, (const float*)nullptr, XR, N);
    hipLaunchKernelGGL(copy_f32, dim3((N * W1F + 255) / 256), B256, 0, stream,
                       AGG, H, N * W1F);              // AGG = identity residual
    hipLaunchKernelGGL(init_softmax, dim3((N * 4 + 255) / 256), B256, 0, stream,
                       NMX, NDN, N * 4);
    hipLaunchKernelGGL(edge_logits4, dim3(edgeBlocks), B256, 0, stream,
                       XL, XR, srcI, dstI, att2, ET, NMX, E, Etot);
    hipLaunchKernelGGL(edge_exp, dim3((Etot * 4 + 255) / 256), B256, 0, stream,
                       dstI, ET, NMX, NDN, E, Etot, 4);
    hipLaunchKernelGGL(edge_agg, dim3(edgeBlocks), B256, 0, stream,
                       XL, srcI, dstI, ET, NDN, AGG, E, Etot, W1F, 4);
    hipLaunchKernelGGL(combine_bn, dim3((N * W1F + 255) / 256), B256, 0, stream,
                       AGG, b2, bn2g, bn2b, bn2m, bn2v, H, N * W1F, W1F, 1);

    // ---------------- Layer 3: 128 -> 47 (heads=1), linear residual --------
    hipLaunchKernelGGL((wmma_gemm_tdm<128, 47>), GEMM_GRID, B128, 0, stream,
                       H, w3s, (const float*)nullptr, XL, N);
    hipLaunchKernelGGL((wmma_gemm_tdm<128, 47>), GEMM_GRID, B128, 0, stream,
                       H, w3d, (const float*)nullptr, XR, N);
    hipLaunchKernelGGL((wmma_gemm_tdm<128, 47>), GEMM_GRID, B128, 0, stream,
                       H, res2w, res2b, AGG, N);      // AGG = residual
    hipLaunchKernelGGL(init_softmax, dim3((N + 255) / 256), B256, 0, stream,
                       NMX, NDN, N);
    hipLaunchKernelGGL(edge_logits1, dim3(edgeBlocks), B256, 0, stream,
                       XL, XR, srcI, dstI, att3, ET, NMX, E, Etot, OUTF);
    hipLaunchKernelGGL(edge_exp, dim3((Etot + 255) / 256), B256, 0, stream,
                       dstI, ET, NMX, NDN, E, Etot, 1);
    hipLaunchKernelGGL(edge_agg, dim3(edgeBlocks), B256, 0, stream,
                       XL, srcI, dstI, ET, NDN, AGG, E, Etot, OUTF, 1);
    hipLaunchKernelGGL(combine_bn, dim3((N * OUTF + 255) / 256), B256, 0, stream,
                       AGG, b3, bn3g, bn3b, bn3m, bn3v, (float*)d_out,
                       N * OUTF, OUTF, 0);
}